// CubicSplineGrid_9517647528174
// MI455X (gfx1250) — compile-verified
//
#include <hip/hip_runtime.h>
#include <stdint.h>

// 4-wide float vector with only 4-byte alignment guarantee: the x-axis taps
// start at an arbitrary float offset, so we must not claim 16B alignment.
typedef float f4u __attribute__((ext_vector_type(4), aligned(4)));
typedef float f2  __attribute__((ext_vector_type(2), aligned(8)));

#define GRID_W   64
#define YSTRIDE  64          // floats
#define ZSTRIDE  4096        // 64*64
#define CSTRIDE  262144      // 64*64*64
#define BLOCK    256

// Catmull-Rom axis: weights remapped onto 4 CONTIGUOUS taps starting at b.
// pos = coord*63, idx = floor(pos), taps = clip(idx-1+k, 0, 63).
// b = clamp(idx-1, 0, 60); s = (idx-1)-b in {-1,0,1,2}; tap k contributes to
// contiguous position clamp(s+k, 0, 3). s==0 for all interior points.
__device__ __forceinline__ void cr_axis(float coord, float wp[4], int& b) {
    float pos  = coord * 63.0f;
    float fidx = floorf(pos);
    float t    = pos - fidx;
    float t2 = t * t;
    float t3 = t2 * t;
    // w = [1,t,t^2,t^3] @ (0.5*[[0,2,0,0],[-1,0,1,0],[2,-5,4,-1],[-1,3,-3,1]])
    float w0 = 0.5f * (2.0f * t2 - t3 - t);
    float w1 = 0.5f * (3.0f * t3 - 5.0f * t2 + 2.0f);
    float w2 = 0.5f * (4.0f * t2 - 3.0f * t3 + t);
    float w3 = 0.5f * (t3 - t2);

    int idx = (int)fidx;
    int im1 = idx - 1;
    b = min(max(im1, 0), GRID_W - 4);
    int s = im1 - b;                       // -1, 0, 1, or 2
    if (__builtin_expect(s == 0, 1)) {     // interior: overwhelmingly common
        wp[0] = w0; wp[1] = w1; wp[2] = w2; wp[3] = w3;
    } else if (s < 0) {                    // idx==0: taps [0,0,1,2]
        wp[0] = w0 + w1; wp[1] = w2; wp[2] = w3; wp[3] = 0.0f;
    } else if (s == 1) {                   // idx==62: taps [61,62,63,63]
        wp[0] = 0.0f; wp[1] = w0; wp[2] = w1; wp[3] = w2 + w3;
    } else {                               // idx==63: taps [62,63,63,63]
        wp[0] = 0.0f; wp[1] = 0.0f; wp[2] = w0; wp[3] = w1 + w2 + w3;
    }
}

__global__ __launch_bounds__(BLOCK) void CubicSplineGrid_9517647528174_kernel(
    const float* __restrict__ grid,   // (2, 64, 64, 64)
    const float* __restrict__ u,      // (N, 3)  [z, y, x] in [0,1)
    float* __restrict__ out,          // (N, 2)
    int npts)
{
    // ---- Stage this block's chunk of u (AoS float3) through LDS with the
    // ---- CDNA5 async global->LDS DMA: fully coalesced b128 per lane instead
    // ---- of per-lane strided b96, no VGPR return slots, tracked by ASYNCcnt.
    __shared__ float s_u[BLOCK * 3];

    const int tid = threadIdx.x;
    const int p0  = blockIdx.x * BLOCK;
    const int nfl = (min(BLOCK, npts - p0)) * 3;      // floats to stage (>0)

    if (tid * 4 + 4 <= nfl) {
        // Full 16B chunk: u base + p0*3 floats is 16B aligned (BLOCK*12 % 16 == 0).
        const float* gsrc = u + (size_t)p0 * 3 + (size_t)tid * 4;
        uint32_t lds_addr = (uint32_t)(uintptr_t)(&s_u[tid * 4]); // low 32b = LDS offset
        asm volatile("global_load_async_to_lds_b128 %0, %1, off"
                     :: "v"(lds_addr), "v"(gsrc)
                     : "memory");
    } else if (tid * 4 < nfl) {
        // Partial tail chunk (last block only): plain copy.
        for (int k = tid * 4; k < nfl; ++k)
            s_u[k] = u[(size_t)p0 * 3 + k];
    }
    // Each wave drains its own async DMA, then the block barrier makes every
    // wave's LDS writes visible to all readers.
    asm volatile("s_wait_asynccnt 0x0" ::: "memory");
    __syncthreads();

    const int n = p0 + tid;
    if (n < npts) {
        // stride-3 LDS reads across lanes: gcd(3,64)=1 -> conflict-free
        float uz = s_u[3 * tid + 0];
        float uy = s_u[3 * tid + 1];
        float ux = s_u[3 * tid + 2];

        float wz[4], wy[4], wx[4];
        int bz, by, bx;
        cr_axis(uz, wz, bz);
        cr_axis(uy, wy, by);
        cr_axis(ux, wx, bx);

        // One 64-bit address; all 32 taps reached via 24-bit immediate offsets
        // (c*1MB + i*16KB + j*256B all < 8MB).
        const float* base = grid + (size_t)bz * ZSTRIDE + (size_t)by * YSTRIDE + bx;

        f2 res;
#pragma unroll
        for (int c = 0; c < 2; ++c) {
            const float* gc = base + (size_t)c * CSTRIDE;
            float vz = 0.0f;
#pragma unroll
            for (int i = 0; i < 4; ++i) {
                float vy = 0.0f;
#pragma unroll
                for (int j = 0; j < 4; ++j) {
                    // 4 contiguous x-taps: one global_load_b128 per (c,i,j),
                    // default (RT) policy keeps the 2MB grid L2-resident.
                    f4u g = *reinterpret_cast<const f4u*>(gc + i * ZSTRIDE + j * YSTRIDE);
                    float vx = g.x * wx[0];
                    vx = fmaf(g.y, wx[1], vx);
                    vx = fmaf(g.z, wx[2], vx);
                    vx = fmaf(g.w, wx[3], vx);
                    vy = fmaf(wy[j], vx, vy);
                }
                vz = fmaf(wz[i], vy, vz);
            }
            res[c] = vz;
        }

        // (N,2) row is 8B-aligned -> single b64 non-temporal streaming store.
        __builtin_nontemporal_store(res, reinterpret_cast<f2*>(out) + n);
    }
}

extern "C" void kernel_launch(void* const* d_in, const int* in_sizes, int n_in,
                              void* d_out, int out_size, void* d_ws, size_t ws_size,
                              hipStream_t stream) {
    const float* grid = (const float*)d_in[0];   // 2*64*64*64 floats
    const float* u    = (const float*)d_in[1];   // N*3 floats
    float* out        = (float*)d_out;           // N*2 floats

    int npts = in_sizes[1] / 3;
    int nblocks = (npts + BLOCK - 1) / BLOCK;
    CubicSplineGrid_9517647528174_kernel<<<nblocks, BLOCK, 0, stream>>>(grid, u, out, npts);
}